// QuerySelector_66597762892385
// MI455X (gfx1250) — compile-verified
//
#include <hip/hip_runtime.h>
#include <stdint.h>

// QuerySelector: scores = tokens([16,20000,256]) . W([256]); top-300 per batch
// (sorted desc, ties -> lowest index, matching jax.lax.top_k); gather rows.
// Bias b does not affect any returned output (pure rank shift) -> ignored.

typedef float v2f __attribute__((ext_vector_type(2)));
typedef float v8f __attribute__((ext_vector_type(8)));

#define DMODEL 256
#define NUM_Q  300

// ---------------------------------------------------------------------------
// Kernel 1: scoring via V_WMMA_F32_16X16X4_F32.
// One wave computes scores for a tile of 16 consecutive token rows.
// A (16x4 fp32): lanes 0-15 = rows, VGPR0/1 = K {0,1} (lanes<16) / {2,3} (lanes>=16)
// B (4x16 fp32): W chunk replicated over all 16 columns -> same per-lane values.
// D (16x16 f32): every column identical == dot(token_m, W); lanes 0 and 16
// each hold 8 finished scores (rows 0-7 and 8-15 of the tile).
// ---------------------------------------------------------------------------
__global__ __launch_bounds__(256) void qs_score_wmma(
    const float* __restrict__ tokens, const float* __restrict__ W,
    float* __restrict__ scores, int n_tiles)
{
  const int wave = blockIdx.x * (blockDim.x >> 5) + (threadIdx.x >> 5);
  if (wave >= n_tiles) return;                 // wave-uniform: EXEC stays full
  const int lane = threadIdx.x & 31;
  const int m    = lane & 15;                  // token row within tile
  const int kh   = (lane >> 4) << 1;           // 0 for lanes<16, 2 for lanes>=16

  const long  row0 = (long)wave * 16;
  const float* __restrict__ arow = tokens + (row0 + m) * DMODEL + kh;
  const float* __restrict__ wrow = W + kh;

  v8f acc = {};
  #pragma unroll 4
  for (int k0 = 0; k0 < DMODEL; k0 += 4) {
    const float2 av = *reinterpret_cast<const float2*>(arow + k0);  // b64 load
    const float2 bv = *reinterpret_cast<const float2*>(wrow + k0);
    v2f a; a.x = av.x; a.y = av.y;
    v2f b; b.x = bv.x; b.y = bv.y;
    acc = __builtin_amdgcn_wmma_f32_16x16x4_f32(
        /*neg_a=*/false, a, /*neg_b=*/false, b,
        /*c_mod=*/(short)0, acc, /*reuse_a=*/false, /*reuse_b=*/false);
  }

  if ((lane & 15) == 0) {                      // lanes 0 (rows 0-7) and 16 (rows 8-15)
    const long base = row0 + ((lane >> 4) << 3);
    #pragma unroll
    for (int j = 0; j < 8; ++j) scores[base + j] = acc[j];
  }
}

// ---------------------------------------------------------------------------
// Kernel 2: exact per-batch top-300 (one block per batch, 1024 threads).
// 4-pass radix-select on order-preserving uint keys, then candidate collect
// + 1024-wide bitonic sort descending on (value desc, index asc).
// ---------------------------------------------------------------------------
__device__ __forceinline__ unsigned sortable_key(float f) {
  unsigned u = __float_as_uint(f);
  return (u & 0x80000000u) ? ~u : (u | 0x80000000u);
}

__global__ __launch_bounds__(1024) void qs_topk(
    const float* __restrict__ scores, int* __restrict__ topk, int N)
{
  const int b   = blockIdx.x;
  const int tid = threadIdx.x;
  const float* __restrict__ s = scores + (long)b * N;

  __shared__ unsigned hist[256];
  __shared__ unsigned long long cand[1024];
  __shared__ int s_sel, s_need, s_cnt;

  // ---- radix select: exact key of the 300th-largest element ----
  unsigned prefix = 0, mask = 0;
  int need = NUM_Q;
  for (int pass = 0; pass < 4; ++pass) {
    const int shift = 24 - 8 * pass;
    for (int v = tid; v < 256; v += 1024) hist[v] = 0;
    __syncthreads();
    for (int i = tid; i < N; i += 1024) {
      const unsigned u = sortable_key(s[i]);
      if ((u & mask) == prefix) atomicAdd(&hist[(u >> shift) & 0xFFu], 1u);
    }
    __syncthreads();
    if (tid == 0) {
      int cum = 0, sel = 0, nn = need;
      for (int v = 255; v >= 0; --v) {
        const int c = (int)hist[v];
        if (cum + c >= need) { sel = v; nn = need - cum; break; }
        cum += c;
      }
      s_sel = sel; s_need = nn;
    }
    __syncthreads();
    prefix |= ((unsigned)s_sel) << shift;
    mask   |= 0xFFu << shift;
    need = s_need;
    __syncthreads();                      // hist reused next pass
  }
  const unsigned thresh = prefix;

  // ---- collect candidates >= threshold ----
  if (tid == 0) s_cnt = 0;
  cand[tid] = 0ull;                       // pad value sorts last
  __syncthreads();
  for (int i = tid; i < N; i += 1024) {
    const unsigned u = sortable_key(s[i]);
    if (u >= thresh) {
      const int p = atomicAdd(&s_cnt, 1);
      if (p < 1024)
        cand[p] = ((unsigned long long)u << 32) | (unsigned)(~(unsigned)i);
    }
  }
  __syncthreads();

  // ---- bitonic sort, 1024 elems, descending (value desc, index asc) ----
  for (int k = 2; k <= 1024; k <<= 1) {
    for (int j = k >> 1; j > 0; j >>= 1) {
      const int ixj = tid ^ j;
      if (ixj > tid) {
        const unsigned long long a = cand[tid];
        const unsigned long long c = cand[ixj];
        const bool desc = ((tid & k) == 0);
        if (desc ? (a < c) : (a > c)) { cand[tid] = c; cand[ixj] = a; }
      }
      __syncthreads();
    }
  }

  if (tid < NUM_Q)
    topk[b * NUM_Q + tid] = (int)(~(unsigned)(cand[tid] & 0xFFFFFFFFull));
}

// ---------------------------------------------------------------------------
// Kernel 3: gather selected token rows (one block per (row, batch)).
// ---------------------------------------------------------------------------
__global__ __launch_bounds__(64) void qs_gather(
    const float* __restrict__ tokens, const int* __restrict__ topk,
    float* __restrict__ selected, int N)
{
  const int r = blockIdx.x;
  const int b = blockIdx.y;
  const int idx = topk[b * NUM_Q + r];
  const float4* __restrict__ src =
      reinterpret_cast<const float4*>(tokens + ((long)b * N + idx) * DMODEL);
  float4* __restrict__ dst =
      reinterpret_cast<float4*>(selected + ((long)b * NUM_Q + r) * DMODEL);
  dst[threadIdx.x] = src[threadIdx.x];
}

// ---------------------------------------------------------------------------
extern "C" void kernel_launch(void* const* d_in, const int* in_sizes, int n_in,
                              void* d_out, int out_size, void* d_ws, size_t ws_size,
                              hipStream_t stream) {
  const float* tokens = (const float*)d_in[0];
  const float* W      = (const float*)d_in[1];
  // d_in[2] = bias (scalar): rank-invariant, not needed for any output.

  const int  d          = in_sizes[1];                 // 256
  const long total_rows = (long)in_sizes[0] / d;       // B * N
  const int  B          = (int)((long)out_size / (NUM_Q * (d + 1)));  // 16
  const int  N          = (int)(total_rows / B);       // 20000

  float* selected = (float*)d_out;
  int*   topk     = (int*)(selected + (long)B * NUM_Q * d);
  float* scores   = (float*)d_ws;                      // B*N floats = 1.28 MB

  const int n_tiles = (int)(total_rows / 16);          // 20000 tiles of 16 rows
  const int waves_per_block = 8;                       // 256 threads
  const int score_blocks = (n_tiles + waves_per_block - 1) / waves_per_block;

  qs_score_wmma<<<dim3(score_blocks), dim3(256), 0, stream>>>(tokens, W, scores, n_tiles);
  qs_topk<<<dim3(B), dim3(1024), 0, stream>>>(scores, topk, N);
  qs_gather<<<dim3(NUM_Q, B), dim3(d / 4), 0, stream>>>(tokens, topk, selected, N);
}